// MemoryRamModule_40389872452048
// MI455X (gfx1250) — compile-verified
//
#include <hip/hip_runtime.h>
#include <hip/hip_bf16.h>
#include <math.h>

// ---------------------------------------------------------------------------
// MemoryRamModule on MI455X (gfx1250)
//
// Phase 0: repack all x-part weights into W2T[1280][1024] (K-contiguous)
// Phase 1: PRE[16384][1280] = X @ W2T^T via V_WMMA_F32_16X16X4_F32 (f32 WMMA,
//          full reference precision). ~41 GFLOP, fully parallel.
// Phase 2: sequential scan, 16 persistent workgroups, one per WGP, with all
//          h-dependent weights resident in LDS (~250KB of 320KB/WGP).
//          3 global barriers per step (minimum for the h->logits->r->h_new
//          dependency chain); cooperative wave-reduced softmax; ds_load_b128
//          bank-conflict-free matvecs; global_prefetch of next PRE row.
// ---------------------------------------------------------------------------

typedef float v2f __attribute__((ext_vector_type(2)));
typedef float v4f __attribute__((ext_vector_type(4)));
typedef float v8f __attribute__((ext_vector_type(8)));

#define I_SZ   1024
#define H_SZ   512
#define MB_SZ  100
#define NSTEP  16384
#define NCOL   1280          // 1226 padded to multiple of 16
#define NWG    16
#define TPB    256
#define ROWS_PER_WG 77       // 16*77 = 1232 >= 1226 stacked h-rows
#define COLS_PER_WG 32       // 16*32 = 512 Wrh output columns
#define MROWS  7             // 16*7  = 112 >= 100 memory rows
#define UR_STRIDE 520        // floats; rows 16B-aligned; b128 conflict-free
#define WRT_STRIDE 520

// workspace float offsets
#define OFF_CTRL 0                   // int barrier counter in slot 0
#define OFF_H    16                  // h double buffer [2][512]
#define OFF_R    (OFF_H + 2*H_SZ)    // r accumulator double buffer [2][512]
#define OFF_C    (OFF_R + 2*H_SZ)    // c broadcast [512]
#define OFF_UHH  (OFF_C + H_SZ)      // u_hh broadcast [512]
#define OFF_HLOG (OFF_UHH + H_SZ)    // h-part logits [256] (202 used)
#define OFF_W2T  4096
#define OFF_PRE  (OFF_W2T + NCOL*I_SZ)
// total ws: (OFF_PRE + NSTEP*NCOL) floats ~= 89.2 MB

#define SCAN_SMEM_FLOATS (ROWS_PER_WG*UR_STRIDE + COLS_PER_WG*WRT_STRIDE + \
                          MROWS*H_SZ + 3*H_SZ + 4*112 + 16)
#define SCAN_SMEM_BYTES  (SCAN_SMEM_FLOATS * 4)

__global__ void init_kernel(float* __restrict__ wsf) {
    int i = blockIdx.x * blockDim.x + threadIdx.x;
    if (i < 4096) wsf[i] = 0.f;   // zeroes barrier counter, h, r, c, uhh, hlog
}

// Gather x-part weights into W2T[n][k] = weight(output n, input k), k contiguous.
__global__ void repack_kernel(const float* __restrict__ Wxh,
                              const float* __restrict__ Whc,
                              const float* __restrict__ Wwp,
                              const float* __restrict__ Wrp,
                              const float* __restrict__ Wwg,
                              const float* __restrict__ Wrg,
                              float* __restrict__ W2T) {
    int idx = blockIdx.x * blockDim.x + threadIdx.x;  // over 1280*1024
    int n = idx >> 10, k = idx & 1023;
    float v = 0.f;
    if (n < 512)        v = Wxh[(size_t)k * H_SZ + n];            // x @ Wxh
    else if (n < 1024)  v = Whc[(size_t)(n - 512)  * 1536 + k];   // Whc x-part
    else if (n < 1124)  v = Wwp[(size_t)(n - 1024) * 1536 + k];
    else if (n < 1224)  v = Wrp[(size_t)(n - 1124) * 1536 + k];
    else if (n == 1224) v = Wwg[k];
    else if (n == 1225) v = Wrg[k];
    W2T[idx] = v;
}

// PRE[m][n] = sum_k X[m][k] * W2T[n][k], one 16x16 tile per wave,
// K accumulated through chained V_WMMA_F32_16X16X4_F32.
__global__ void __launch_bounds__(256) gemm_kernel(const float* __restrict__ X,
                                                   const float* __restrict__ W2T,
                                                   float* __restrict__ PRE) {
    const int wave = (blockIdx.x * blockDim.x + threadIdx.x) >> 5;
    const int lane = threadIdx.x & 31;
    const int mt = wave / (NCOL / 16);
    const int nt = wave % (NCOL / 16);
    const int m0 = mt << 4, n0 = nt << 4;
    const int half = lane >> 4, lr = lane & 15;

    const float* xrow = X   + (size_t)(m0 + lr) * I_SZ;  // A row M=lr
    const float* wrow = W2T + (size_t)(n0 + lr) * I_SZ;  // B col N=lr
    v8f acc = {};
    #pragma unroll 8
    for (int k0 = 0; k0 < I_SZ; k0 += 4) {
        // f32 A 16x4 layout: lanes<16 hold K=k0,k0+1 ; lanes>=16 hold K=k0+2,k0+3
        v2f a = *(const v2f*)(xrow + k0 + 2 * half);
        v2f b = *(const v2f*)(wrow + k0 + 2 * half);
        acc = __builtin_amdgcn_wmma_f32_16x16x4_f32(
            false, a, false, b, (short)0, acc, false, false);
    }
    // D layout: vgpr r, lanes<16 -> M=r, lanes>=16 -> M=8+r
    float* o = PRE + (size_t)(m0 + 8 * half) * NCOL + n0 + lr;
    #pragma unroll
    for (int r = 0; r < 8; ++r) o[(size_t)r * NCOL] = acc[r];
}

__device__ __forceinline__ void gbar(int* cnt, int target) {
    __threadfence();          // release my stores to device scope
    __syncthreads();
    if (threadIdx.x == 0) {
        atomicAdd(cnt, 1);
        while (__hip_atomic_load(cnt, __ATOMIC_RELAXED,
                                 __HIP_MEMORY_SCOPE_AGENT) < target) {
            __builtin_amdgcn_s_sleep(1);
        }
    }
    __syncthreads();
    __threadfence();          // acquire peers' stores
}

__device__ __forceinline__ float sigmoidf_(float x) {
    return 1.f / (1.f + expf(-x));
}

__global__ void __launch_bounds__(TPB, 1) scan_kernel(
    const float* __restrict__ PRE,
    const float* __restrict__ Whc, const float* __restrict__ bhc,
    const float* __restrict__ Wwg, const float* __restrict__ bwg,
    const float* __restrict__ Wwp, const float* __restrict__ bwp,
    const float* __restrict__ Wrg, const float* __restrict__ brg,
    const float* __restrict__ Wrp, const float* __restrict__ brp,
    const float* __restrict__ Wrh, const float* __restrict__ Whh,
    const float* __restrict__ bh,
    float* __restrict__ h_buf, float* __restrict__ r_acc,
    float* __restrict__ c_buf, float* __restrict__ uhh,
    float* __restrict__ hlog, int* __restrict__ cnt,
    float* __restrict__ out) {
    extern __shared__ float sm[];
    float* Ur   = sm;                                 // [77][520]
    float* WrT  = Ur  + ROWS_PER_WG * UR_STRIDE;      // [32][520]
    float* Ms   = WrT + COLS_PER_WG * WRT_STRIDE;     // [7][512]
    float* hs   = Ms  + MROWS * H_SZ;                 // [512]
    float* rs   = hs  + H_SZ;                         // [512]
    float* cs   = rs  + H_SZ;                         // [512]
    float* lrp  = cs  + H_SZ;                         // [112] read logits
    float* lwp  = lrp + 112;                          // [112] write logits
    float* ar_s = lwp + 112;                          // [112] read attention
    float* aw_s = ar_s + 112;                         // [112] write attention
    float* stat = aw_s + 112;  // [16]: 0=mrp 1=srp 2=mwp 3=swp 4=go_l 5=gw_l

    const int g = blockIdx.x;
    const int tid = threadIdx.x;

    // Load resident weight slices once. Stacked h-row R:
    // [0,512): Whh col R | [512,1024): Whc_h | [1024,1124): Wwp_h
    // [1124,1224): Wrp_h | 1224: Wwg_h | 1225: Wrg_h
    for (int idx = tid; idx < ROWS_PER_WG * H_SZ; idx += TPB) {
        int r = idx >> 9, k = idx & 511;
        int R = g * ROWS_PER_WG + r;
        float v = 0.f;
        if (R < 512)        v = Whh[(size_t)k * H_SZ + R];
        else if (R < 1024)  v = Whc[(size_t)(R - 512)  * 1536 + 1024 + k];
        else if (R < 1124)  v = Wwp[(size_t)(R - 1024) * 1536 + 1024 + k];
        else if (R < 1224)  v = Wrp[(size_t)(R - 1124) * 1536 + 1024 + k];
        else if (R == 1224) v = Wwg[1024 + k];
        else if (R == 1225) v = Wrg[1024 + k];
        Ur[r * UR_STRIDE + k] = v;
    }
    for (int idx = tid; idx < COLS_PER_WG * H_SZ; idx += TPB) {
        int c = idx >> 9, k = idx & 511;
        WrT[c * WRT_STRIDE + k] = Wrh[(size_t)k * H_SZ + (g * COLS_PER_WG + c)];
    }
    for (int idx = tid; idx < MROWS * H_SZ; idx += TPB) Ms[idx] = 0.f;  // M0 = 0
    if (tid >= MB_SZ && tid < 112) { ar_s[tid] = 0.f; aw_s[tid] = 0.f; } // pads
    __syncthreads();

    const v4f* hs4 = (const v4f*)hs;
    const v4f* rs4 = (const v4f*)rs;

    int ev = 0;
    for (int t = 0; t < NSTEP; ++t) {
        const int buf = t & 1;
        const float* pre = PRE + (size_t)t * NCOL;

        for (int k = tid; k < H_SZ; k += TPB) hs[k] = h_buf[buf * H_SZ + k];
        // prefetch next step's x-part row (5KB) behind the upcoming barriers
        if (t + 1 < NSTEP && tid < 40)
            __builtin_prefetch(PRE + (size_t)(t + 1) * NCOL + tid * 32, 0, 1);
        __syncthreads();

        // u = Ur @ h ; 4 lanes per row; interleaved float4 (b128, no conflicts)
        {
            const int seg = tid & 3;
            for (int rr = (tid >> 2); rr < ROWS_PER_WG; rr += (TPB >> 2)) {
                const v4f* w4 = (const v4f*)&Ur[rr * UR_STRIDE];
                float acc = 0.f;
                #pragma unroll 4
                for (int i = 0; i < 32; ++i) {
                    const int kq = seg + (i << 2);
                    v4f wv = w4[kq], hv = hs4[kq];
                    acc += wv[0] * hv[0] + wv[1] * hv[1] +
                           wv[2] * hv[2] + wv[3] * hv[3];
                }
                acc += __shfl_xor(acc, 1);
                acc += __shfl_xor(acc, 2);
                int R = g * ROWS_PER_WG + rr;
                if (seg == 0 && R < 1226) {
                    if (R < 512) uhh[R] = acc;
                    else if (R < 1024) {
                        int j = R - 512;
                        c_buf[j] = fmaxf(pre[512 + j] + acc + bhc[j], 0.f);
                    } else hlog[R - 1024] = acc;
                }
            }
        }
        ev++; gbar(cnt, ev * NWG);   // B1: logits / uhh / c published

        // ---- cooperative softmax/gates: stage logits in LDS ----
        if (tid < MB_SZ) {
            lrp[tid] = pre[1124 + tid] + hlog[100 + tid] + brp[tid];
            lwp[tid] = pre[1024 + tid] + hlog[tid]       + bwp[tid];
        } else if (tid == 100) {
            stat[4] = pre[1225] + hlog[201] + brg[0];    // read-gate logit
        } else if (tid == 101) {
            stat[5] = pre[1224] + hlog[200] + bwg[0];    // write-gate logit
        }
        __syncthreads();
        if (tid < 64) {               // wave0: rp, wave1: wp
            const float* lg = (tid < 32) ? lrp : lwp;
            const int lane = tid & 31;
            float m = -3.4e38f;
            for (int i = lane; i < MB_SZ; i += 32) m = fmaxf(m, lg[i]);
            #pragma unroll
            for (int o = 16; o; o >>= 1) m = fmaxf(m, __shfl_xor(m, o));
            float s = 0.f;
            for (int i = lane; i < MB_SZ; i += 32) s += expf(lg[i] - m);
            #pragma unroll
            for (int o = 16; o; o >>= 1) s += __shfl_xor(s, o);
            if (lane == 0) { stat[(tid < 32) ? 0 : 2] = m;
                             stat[(tid < 32) ? 1 : 3] = s; }
        }
        __syncthreads();
        if (tid < MB_SZ) {
            ar_s[tid] = expf(lrp[tid] - stat[0]) / stat[1];
            aw_s[tid] = expf(lwp[tid] - stat[2]) / stat[3];
        }
        __syncthreads();
        const float go = sigmoidf_(stat[4]);
        const float gw = sigmoidf_(stat[5]);

        // partial r = go * (ar_slice @ M_slice), accumulated device-wide
        for (int k = tid; k < H_SZ; k += TPB) {
            float acc = 0.f;
            #pragma unroll
            for (int i = 0; i < MROWS; ++i)
                acc += ar_s[g * MROWS + i] * Ms[(i << 9) + k];
            atomicAdd(&r_acc[buf * H_SZ + k], go * acc);
        }
        ev++; gbar(cnt, ev * NWG);   // B2: r complete

        for (int k = tid; k < H_SZ; k += TPB) rs[k] = r_acc[buf * H_SZ + k];
        for (int k = tid; k < H_SZ; k += TPB) cs[k] = c_buf[k];
        __syncthreads();

        // h_new_j = relu(x@Wxh + r@Wrh + h@Whh + bh) ; 8 lanes per column
        {
            const int c = tid >> 3, seg = tid & 7;
            const int j = g * COLS_PER_WG + c;
            const v4f* w4 = (const v4f*)&WrT[c * WRT_STRIDE];
            float acc = 0.f;
            #pragma unroll 4
            for (int i = 0; i < 16; ++i) {
                const int kq = seg + (i << 3);
                v4f wv = w4[kq], rv = rs4[kq];
                acc += wv[0] * rv[0] + wv[1] * rv[1] +
                       wv[2] * rv[2] + wv[3] * rv[3];
            }
            acc += __shfl_xor(acc, 1);
            acc += __shfl_xor(acc, 2);
            acc += __shfl_xor(acc, 4);
            if (seg == 0) {
                float hn = fmaxf(pre[j] + uhh[j] + bh[j] + acc, 0.f);
                h_buf[(1 - buf) * H_SZ + j] = hn;
                out[(size_t)t * H_SZ + j] = hn;
            }
        }

        // memory bank update on owned rows (aw padded with zeros -> no guard)
        for (int idx = tid; idx < MROWS * H_SZ; idx += TPB) {
            int i = idx >> 9, k = idx & 511;
            float a = aw_s[g * MROWS + i];
            Ms[idx] = gw * a * cs[k] + (1.f - a) * Ms[idx];
        }
        if (g == 0)   // zero next step's r accumulator (double buffered)
            for (int k = tid; k < H_SZ; k += TPB) r_acc[(1 - buf) * H_SZ + k] = 0.f;
        ev++; gbar(cnt, ev * NWG);   // B3: step complete
    }
}

extern "C" void kernel_launch(void* const* d_in, const int* in_sizes, int n_in,
                              void* d_out, int out_size, void* d_ws, size_t ws_size,
                              hipStream_t stream) {
    const float* X   = (const float*)d_in[0];
    const float* Whc = (const float*)d_in[1];
    const float* bhc = (const float*)d_in[2];
    const float* Wwg = (const float*)d_in[3];
    const float* bwg = (const float*)d_in[4];
    const float* Wwp = (const float*)d_in[5];
    const float* bwp = (const float*)d_in[6];
    const float* Wrg = (const float*)d_in[7];
    const float* brg = (const float*)d_in[8];
    const float* Wrp = (const float*)d_in[9];
    const float* brp = (const float*)d_in[10];
    const float* Wxh = (const float*)d_in[11];
    const float* Wrh = (const float*)d_in[12];
    const float* Whh = (const float*)d_in[13];
    const float* bh  = (const float*)d_in[14];
    float* out = (float*)d_out;
    float* wsf = (float*)d_ws;

    float* W2T   = wsf + OFF_W2T;
    float* PRE   = wsf + OFF_PRE;
    float* h_buf = wsf + OFF_H;
    float* r_acc = wsf + OFF_R;
    float* c_buf = wsf + OFF_C;
    float* uhh   = wsf + OFF_UHH;
    float* hlog  = wsf + OFF_HLOG;
    int*   cnt   = (int*)wsf;

    (void)hipFuncSetAttribute(reinterpret_cast<const void*>(scan_kernel),
                              hipFuncAttributeMaxDynamicSharedMemorySize,
                              SCAN_SMEM_BYTES);

    init_kernel<<<4, 1024, 0, stream>>>(wsf);
    repack_kernel<<<(NCOL * I_SZ) / 256, 256, 0, stream>>>(
        Wxh, Whc, Wwp, Wrp, Wwg, Wrg, W2T);
    // (NSTEP/16)*(NCOL/16) tiles, one per wave, 8 waves per block
    gemm_kernel<<<(NSTEP / 16) * (NCOL / 16) / 8, 256, 0, stream>>>(X, W2T, PRE);
    scan_kernel<<<NWG, TPB, SCAN_SMEM_BYTES, stream>>>(
        PRE, Whc, bhc, Wwg, bwg, Wwp, bwp, Wrg, brg, Wrp, brp,
        Wrh, Whh, bh, h_buf, r_acc, c_buf, uhh, hlog, cnt, out);
}